// DiscriminativeLoss_3401614098830
// MI455X (gfx1250) — compile-verified
//
#include <hip/hip_runtime.h>
#include <math.h>

typedef float v2f __attribute__((ext_vector_type(2)));
typedef float v8f __attribute__((ext_vector_type(8)));
typedef int   v4i_t __attribute__((ext_vector_type(4)));
typedef int   v8i_t __attribute__((ext_vector_type(8)));

#define K_CLUSTERS 16
#define DIMS 32
#define DELTA_VAR 0.5f
#define DELTA_DIST 1.5f

// Workspace layout (floats):
#define WS_SUMS    0      // [16][32]
#define WS_COUNTS  512    // [16]
#define WS_CENTERS 528    // [16][32]
#define WS_VAR     1040   // scalar accumulator
#define WS_CONST   1041   // dist + reg terms (weighted)
#define WS_TOTAL   1042

__global__ void k_init(float* ws) {
  int i = blockIdx.x * blockDim.x + threadIdx.x;
  if (i < WS_TOTAL) ws[i] = 0.0f;
}

// ---------------------------------------------------------------------------
// Pass 1: per-cluster sums & counts via WMMA:  sums[16x32] = onehot[16xN] * pts[Nx32]
// Per wave: 16-point tile. A = one-hot (from labels via shuffles, no loads),
// B = pts tile DMA'd into LDS by the Tensor Data Mover (one tensor_load_to_lds
// per tile; TDM pad feature gives the pitch-17 conflict-free layout directly),
// C accumulated in registers across all tiles; flushed once via LDS + global atomics.
// ---------------------------------------------------------------------------
__global__ __launch_bounds__(256) void k_segsum(const float* __restrict__ data,
                                                const int* __restrict__ labels,
                                                float* __restrict__ ws,
                                                int N, int tiles) {
  __shared__ float tileBuf[8][17 * 32];   // per-wave DMA tile [d][point], pitch 17
  __shared__ float s_acc[512];
  __shared__ float s_cnt[16];
  const int tid = threadIdx.x;
  for (int i = tid; i < 512; i += 256) s_acc[i] = 0.0f;
  if (tid < 16) s_cnt[tid] = 0.0f;
  __syncthreads();

  const int lane = tid & 31;
  const int wv   = tid >> 5;
  const int h    = lane >> 4;    // lane half: selects K pair / row group
  const int j    = lane & 15;    // M row (cluster) for A, N col (dim) for B
  const int gw   = blockIdx.x * 8 + wv;
  const int W    = gridDim.x * 8;
  float* lds = tileBuf[wv];

  // Wave-relative LDS byte address of this wave's tile (flat addr low 32 bits).
  const unsigned ldsBase =
      (unsigned)__builtin_amdgcn_readfirstlane((int)(unsigned)(size_t)(void*)lds);

  // ---- TDM descriptor Group 1 (loop-invariant) ------------------------------
  // [17:16] data_size=2 (4B)  [20] pad_enable=1
  // [24:22] pad_interval=3 (every 16 DWORDs)  [31:25] pad_amount=0 (1 DWORD) -> pitch 17
  // tensor_dim0=N @ [79:48], tensor_dim1=32 @ [111:80],
  // tile_dim0=16 @ [127:112], tile_dim1=32 @ [143:128], tile_dim2=0,
  // tensor_dim0_stride=N @ [207:160], tensor_dim1_stride=0.
  v8i_t g1;
  g1[0] = (2 << 16) | (1 << 20) | (3 << 22);
  g1[1] = (int)(((unsigned)N & 0xffffu) << 16);
  g1[2] = (int)((((unsigned)N >> 16) & 0xffffu) | (32u << 16));
  g1[3] = (int)(16u << 16);
  g1[4] = 32;
  g1[5] = N;
  g1[6] = 0;
  g1[7] = 0;

  v8f c0 = {};   // sums tile, dims 0..15   (row = cluster 8h+v, col = dim j)
  v8f c1 = {};   // sums tile, dims 16..31
  float cnt = 0.0f;

  for (int t = gw; t < tiles; t += W) {
    const int base  = t * 16;
    const int ubase = __builtin_amdgcn_readfirstlane(base);
    // ---- TDM descriptor Group 0: count=1, lds_addr, 57-bit global_addr, type=2
    const unsigned long long gaddr =
        (unsigned long long)(size_t)data + (unsigned long long)(unsigned)ubase * 4ull;
    v4i_t g0;
    g0[0] = 1;
    g0[1] = (int)ldsBase;
    g0[2] = (int)(unsigned)(gaddr & 0xffffffffull);
    g0[3] = (int)(unsigned)(((gaddr >> 32) & 0x1ffffffull) | 0x80000000u); // type=2
    // WAR guard: previous tile's ds_loads must finish before DMA overwrites LDS.
    asm volatile("s_wait_dscnt 0x0\n\t"
                 "tensor_load_to_lds %0, %1"
                 :: "s"(g0), "s"(g1) : "memory");
    __builtin_amdgcn_s_wait_tensorcnt(0);
    __builtin_amdgcn_wave_barrier();

    const int lab = labels[base + j];   // lane j holds label of point base+j
#pragma unroll
    for (int g = 0; g < 4; ++g) {
      const int p0 = 4 * g + 2 * h;                 // K index pair for this lane half
      const int l0 = __shfl(lab, p0, 32);
      const int l1 = __shfl(lab, p0 + 1, 32);
      v2f a; a.x = (l0 == j) ? 1.0f : 0.0f;         // one-hot A[cluster j][point p]
      a.y = (l1 == j) ? 1.0f : 0.0f;
      cnt += a.x + a.y;                              // per-cluster count, per half
      v2f b0, b1;                                    // B[point p][dim j]
      b0.x = lds[j * 17 + p0];        b0.y = lds[j * 17 + p0 + 1];
      b1.x = lds[(16 + j) * 17 + p0]; b1.y = lds[(16 + j) * 17 + p0 + 1];
      c0 = __builtin_amdgcn_wmma_f32_16x16x4_f32(false, a, false, b0, (short)0, c0, false, false);
      c1 = __builtin_amdgcn_wmma_f32_16x16x4_f32(false, a, false, b1, (short)0, c1, false, false);
    }
    __builtin_amdgcn_wave_barrier();
  }

  // Flush C (row = cluster 8h+v, col = dim j) into block accumulator.
#pragma unroll
  for (int v = 0; v < 8; ++v) {
    const int row = 8 * h + v;
    atomicAdd(&s_acc[row * 32 + j],      c0[v]);
    atomicAdd(&s_acc[row * 32 + 16 + j], c1[v]);
  }
  float cf = cnt + __shfl_xor(cnt, 16, 32);
  if (lane < 16) atomicAdd(&s_cnt[j], cf);
  __syncthreads();
  for (int i = tid; i < 512; i += 256) atomicAdd(&ws[WS_SUMS + i], s_acc[i]);
  if (tid < 16) atomicAdd(&ws[WS_COUNTS + tid], s_cnt[tid]);
}

// ---------------------------------------------------------------------------
// Centers + pairwise distance term + regularizer (K=16, tiny -> one block)
// ---------------------------------------------------------------------------
__global__ __launch_bounds__(256) void k_centers(float* __restrict__ ws) {
  const int tid = threadIdx.x;
  __shared__ float sc[512];
  __shared__ float red[256];
  __shared__ float red2[16];
  for (int i = tid; i < 512; i += 256) {
    float cval = ws[WS_SUMS + i] / ws[WS_COUNTS + i / 32];
    ws[WS_CENTERS + i] = cval;
    sc[i] = cval;
  }
  __syncthreads();
  const int ci = tid >> 4, cj = tid & 15;
  float hinge = 0.0f;
  if (ci != cj) {
    float s = 0.0f;
#pragma unroll
    for (int d = 0; d < 32; ++d) { float df = sc[ci * 32 + d] - sc[cj * 32 + d]; s += df * df; }
    float pd = sqrtf(s);
    float hh = fmaxf(2.0f * DELTA_DIST - pd, 0.0f);
    hinge = hh * hh;
  }
  red[tid] = hinge;
  if (tid < 16) {
    float s = 0.0f;
#pragma unroll
    for (int d = 0; d < 32; ++d) { float cv = sc[tid * 32 + d]; s += cv * cv; }
    red2[tid] = sqrtf(s);
  }
  __syncthreads();
  for (int off = 128; off > 0; off >>= 1) {
    if (tid < off) red[tid] += red[tid + off];
    __syncthreads();
  }
  if (tid == 0) {
    float regs = 0.0f;
    for (int k = 0; k < 16; ++k) regs += red2[k];
    ws[WS_CONST] = red[0] / (float)(K_CLUSTERS * (K_CLUSTERS - 1))
                 + 0.001f * regs / (float)K_CLUSTERS;
  }
}

// ---------------------------------------------------------------------------
// Pass 2: variance term. dots[16 pts x 16 clusters] = pts[16x32] * centersT[32x16]
// via 8 x v_wmma_f32_16x16x4_f32; ||x||^2, ||c||^2 from already-held registers;
// select dot[p][lab[p]] from the documented C layout with ds_bpermute shuffles.
// ---------------------------------------------------------------------------
__global__ __launch_bounds__(256) void k_var(const float* __restrict__ data,
                                             const int* __restrict__ labels,
                                             const float* __restrict__ ws,
                                             float* __restrict__ var_acc,
                                             int N, int tiles) {
  const int tid  = threadIdx.x;
  const int lane = tid & 31;
  const int h    = lane >> 4;
  const int j    = lane & 15;                 // A: point row; B: cluster col
  const int gw   = blockIdx.x * (blockDim.x >> 5) + (tid >> 5);
  const int W    = gridDim.x * (blockDim.x >> 5);
  const float* centers = ws + WS_CENTERS;

  // Preload B = centersT: lane (h, cluster j), vgpr v of step s holds centers[j][4s+2h+v]
  v2f bc[8];
  float cp = 0.0f;
#pragma unroll
  for (int s = 0; s < 8; ++s) {
    bc[s].x = centers[j * 32 + 4 * s + 2 * h];
    bc[s].y = centers[j * 32 + 4 * s + 2 * h + 1];
    cp += bc[s].x * bc[s].x + bc[s].y * bc[s].y;
  }
  const float cn = cp + __shfl_xor(cp, 16, 32);   // lane (*, j) = ||center_j||^2

  float acc = 0.0f;
  for (int t = gw; t < tiles; t += W) {
    const int base = t * 16;
    const int lab  = labels[base + j];
    v8f c = {};
    float np = 0.0f;
    const bool pf = (t + W) < tiles;
#pragma unroll
    for (int s = 0; s < 8; ++s) {
      const int d0 = 4 * s + 2 * h;
      v2f a;                                          // A[point j][dim d0..d0+1]
      a.x = data[(size_t)d0 * N + base + j];          // coalesced per lane half
      a.y = data[(size_t)(d0 + 1) * N + base + j];
      if (pf) __builtin_prefetch(&data[(size_t)d0 * N + (size_t)(t + W) * 16 + j], 0, 0);
      np += a.x * a.x + a.y * a.y;
      c = __builtin_amdgcn_wmma_f32_16x16x4_f32(false, a, false, bc[s], (short)0, c, false, false);
    }
    const float nf = np + __shfl_xor(np, 16, 32);     // lane (*, j) = ||pt_j||^2
#pragma unroll
    for (int v = 0; v < 8; ++v) {
      const int p     = 8 * h + v;                    // point handled by this half/vgpr
      const int labp  = __shfl(lab, p, 32);
      const float dsel = __shfl(c[v], 16 * h + labp, 32);  // dot[p][lab[p]]
      const float npv  = __shfl(nf, p, 32);
      const float cnv  = __shfl(cn, labp, 32);
      float d2 = fmaxf(cnv - 2.0f * dsel + npv, 0.0f);
      float r  = sqrtf(d2);
      float tt = fmaxf(r - DELTA_VAR, 0.0f);
      acc += tt * tt * 0.0625f;   // 16 lanes per half duplicate each point -> /16
    }
  }
#pragma unroll
  for (int off = 16; off > 0; off >>= 1) acc += __shfl_xor(acc, off, 32);
  if (lane == 0) atomicAdd(var_acc, acc);
}

__global__ void k_final(const float* __restrict__ ws, float* __restrict__ out) {
  if (threadIdx.x == 0 && blockIdx.x == 0)
    out[0] = ws[WS_VAR] / (float)K_CLUSTERS + ws[WS_CONST];
}

extern "C" void kernel_launch(void* const* d_in, const int* in_sizes, int n_in,
                              void* d_out, int out_size, void* d_ws, size_t ws_size,
                              hipStream_t stream) {
  const float* data   = (const float*)d_in[0];
  const int*   labels = (const int*)d_in[1];
  float* ws  = (float*)d_ws;
  float* out = (float*)d_out;
  const int N     = in_sizes[1];   // 512*1024 pixels
  const int tiles = N / 16;

  k_init   <<<(WS_TOTAL + 255) / 256, 256, 0, stream>>>(ws);
  k_segsum <<<128, 256, 0, stream>>>(data, labels, ws, N, tiles);
  k_centers<<<1, 256, 0, stream>>>(ws);
  k_var    <<<256, 256, 0, stream>>>(data, labels, ws, ws + WS_VAR, N, tiles);
  k_final  <<<1, 32, 0, stream>>>(ws, out);
}